// PIOUloss_17051020165141
// MI455X (gfx1250) — compile-verified
//
#include <hip/hip_runtime.h>
#include <math.h>
#include <stdint.h>

#define EPS 1e-8f
#define BLK 256

// 16-byte int vector type matching the builtin's parameter type
// ('__attribute__((__vector_size__(4 * sizeof(int)))) int').
typedef int v4i_ __attribute__((vector_size(16)));

// ---------- CDNA5 async global->LDS staging (ASYNCcnt path) ----------
__device__ __forceinline__ void async_copy_b128(const float* g, float* l) {
#if defined(__has_builtin) && __has_builtin(__builtin_amdgcn_global_load_async_to_lds_b128)
  auto gp = (__attribute__((address_space(1))) v4i_*)(uintptr_t)g;
  auto lp = (__attribute__((address_space(3))) v4i_*)(uint32_t)(uintptr_t)l;
  __builtin_amdgcn_global_load_async_to_lds_b128(gp, lp, 0, 0);
#else
  asm volatile("global_load_async_to_lds_b128 %0, %1, off"
               :: "v"((uint32_t)(uintptr_t)l), "v"(g) : "memory");
#endif
}

__device__ __forceinline__ void wait_async_zero() {
#if defined(__has_builtin) && __has_builtin(__builtin_amdgcn_s_wait_asynccnt)
  __builtin_amdgcn_s_wait_asynccnt(0);
#else
  asm volatile("s_wait_asynccnt 0" ::: "memory");
#endif
}

// ---------- math helpers ----------
__device__ __forceinline__ float nan0(float v) { return (v != v) ? 0.0f : v; }

__device__ __forceinline__ void decode_box(const float* b, float* o) {
  float x = b[0], y = b[1];
  float W = fabsf(b[2]), H = fabsf(b[3]);
  float R1 = fminf(fmaxf(b[4], 0.001f), 0.999f);
  float R2 = fminf(fmaxf(b[5], 0.001f), 0.999f);
  float a1 = W * R1, a2 = H * R2;
  float b1 = W * (1.0f - R1), b2 = H * (1.0f - R2);
  float w1 = sqrtf(a1 * a1 + a2 * a2);
  float w2 = sqrtf(b1 * b1 + b2 * b2);
  float t1 = 1.57079632679489662f + atanf(-a2 / a1);
  float t2 = atanf(b2 / b1);
  float th = (w1 >= w2) ? t1 : t2;
  o[0] = nan0(x); o[1] = nan0(y); o[2] = nan0(w2); o[3] = nan0(w1); o[4] = nan0(th);
}

__device__ __forceinline__ void box2corners(const float* bo, float* cx, float* cy) {
  float x = bo[0], y = bo[1], w = bo[2], h = bo[3], a = bo[4];
  float s, c;
  sincosf(a, &s, &c);
  const float fx[4] = {0.5f, -0.5f, -0.5f, 0.5f};
  const float fy[4] = {0.5f, 0.5f, -0.5f, -0.5f};
#pragma unroll
  for (int k = 0; k < 4; k++) {
    float px = fx[k] * w, py = fy[k] * h;
    cx[k] = px * c - py * s + x;
    cy[k] = px * s + py * c + y;
  }
}

// corners (px,py) tested against box with corners (qx,qy): a=q0, b=q1, d=q3
__device__ __forceinline__ void inbox(const float* px, const float* py,
                                      const float* qx, const float* qy, bool* m) {
  float ax = qx[0], ay = qy[0];
  float abx = qx[1] - ax, aby = qy[1] - ay;
  float adx = qx[3] - ax, ady = qy[3] - ay;
  float nab = abx * abx + aby * aby;
  float nad = adx * adx + ady * ady;
#pragma unroll
  for (int k = 0; k < 4; k++) {
    float amx = px[k] - ax, amy = py[k] - ay;
    float rab = (abx * amx + aby * amy) / nab;
    float rad = (adx * amx + ady * amy) / nad;
    m[k] = (rab > -1e-6f) && (rab < 1.0f + 1e-6f) &&
           (rad > -1e-6f) && (rad < 1.0f + 1e-6f);
  }
}

// ---------- kernel ----------
__global__ __launch_bounds__(BLK)
void piou_kernel(const float* __restrict__ pred, const float* __restrict__ tgt,
                 float* __restrict__ out, int n) {
  __shared__ float sp[BLK * 8];
  __shared__ float st[BLK * 6];
  const int tid = threadIdx.x;
  const long long bstart = (long long)blockIdx.x * BLK;
  const long long i = bstart + tid;

  float p[8], t6[6];
  const bool full = (bstart + BLK <= (long long)n);  // block-uniform
  if (full) {
    const float* gp = pred + bstart * 8;  // 2048 floats = 512 x 16B
    const float* gt = tgt + bstart * 6;   // 1536 floats = 384 x 16B
    async_copy_b128(gp + 4 * tid, &sp[4 * tid]);
    async_copy_b128(gp + 4 * (tid + BLK), &sp[4 * (tid + BLK)]);
    async_copy_b128(gt + 4 * tid, &st[4 * tid]);
    if (tid < BLK / 2)
      async_copy_b128(gt + 4 * (tid + BLK), &st[4 * (tid + BLK)]);
    wait_async_zero();
    __syncthreads();
#pragma unroll
    for (int k = 0; k < 8; k++) p[k] = sp[tid * 8 + k];
#pragma unroll
    for (int k = 0; k < 6; k++) t6[k] = st[tid * 6 + k];
  } else {
    if (i >= n) return;
#pragma unroll
    for (int k = 0; k < 8; k++) p[k] = pred[i * 8 + k];
#pragma unroll
    for (int k = 0; k < 6; k++) t6[k] = tgt[i * 6 + k];
  }

  // ---- decode + corners (c1 = gt, c2 = pred, matching reference) ----
  float p5[5], g5[5];
  decode_box(p, p5);
  decode_box(t6, g5);
  float c1x[4], c1y[4], c2x[4], c2y[4];
  box2corners(g5, c1x, c1y);
  box2corners(p5, c2x, c2y);

  // ---- axis-aligned enclosing boxes ----
  float pminx = fminf(fminf(c2x[0], c2x[1]), fminf(c2x[2], c2x[3]));
  float pminy = fminf(fminf(c2y[0], c2y[1]), fminf(c2y[2], c2y[3]));
  float pmaxx = fmaxf(fmaxf(c2x[0], c2x[1]), fmaxf(c2x[2], c2x[3]));
  float pmaxy = fmaxf(fmaxf(c2y[0], c2y[1]), fmaxf(c2y[2], c2y[3]));
  float gminx = fminf(fminf(c1x[0], c1x[1]), fminf(c1x[2], c1x[3]));
  float gminy = fminf(fminf(c1y[0], c1y[1]), fminf(c1y[2], c1y[3]));
  float gmaxx = fmaxf(fmaxf(c1x[0], c1x[1]), fmaxf(c1x[2], c1x[3]));
  float gmaxy = fmaxf(fmaxf(c1y[0], c1y[1]), fmaxf(c1y[2], c1y[3]));
  float tlx = fmaxf(pminx, gminx), tly = fmaxf(pminy, gminy);
  float brx = fminf(pmaxx, gmaxx), bry = fminf(pmaxy, gmaxy);

  // ---- candidate vertices: 4 gt corners, 4 pred corners, 16 intersections ----
  float vx[24], vy[24];
  bool msk[24];
#pragma unroll
  for (int k = 0; k < 4; k++) { vx[k] = c1x[k]; vy[k] = c1y[k]; }
#pragma unroll
  for (int k = 0; k < 4; k++) { vx[4 + k] = c2x[k]; vy[4 + k] = c2y[k]; }
  inbox(c1x, c1y, c2x, c2y, &msk[0]);
  inbox(c2x, c2y, c1x, c1y, &msk[4]);

#pragma unroll
  for (int a = 0; a < 4; a++) {
    float x1 = c1x[a], y1 = c1y[a];
    float x2 = c1x[(a + 1) & 3], y2 = c1y[(a + 1) & 3];
    float dx1 = x1 - x2, dy1 = y1 - y2;
#pragma unroll
    for (int b = 0; b < 4; b++) {
      float x3 = c2x[b], y3 = c2y[b];
      float x4 = c2x[(b + 1) & 3], y4 = c2y[(b + 1) & 3];
      float dx2 = x3 - x4, dy2 = y3 - y4;
      float num = dx1 * dy2 - dy1 * dx2;
      float den_t = (x1 - x3) * dy2 - (y1 - y3) * dx2;
      bool zero = (num == 0.0f);
      float tpar = zero ? -1.0f : den_t / num;
      bool mt = (tpar > 0.0f) && (tpar < 1.0f);
      float den_u = dx1 * (y1 - y3) - dy1 * (x1 - x3);
      float upar = zero ? -1.0f : -den_u / num;
      bool mu = (upar > 0.0f) && (upar < 1.0f);
      bool m = mt && mu;
      float ts = den_t / (num + EPS);
      float ix = x1 + ts * (x2 - x1);
      float iy = y1 + ts * (y2 - y1);
      int e = 8 + a * 4 + b;
      vx[e] = m ? ix : 0.0f;
      vy[e] = m ? iy : 0.0f;
      msk[e] = m;
    }
  }

  // ---- mean of active vertices ----
  int nv = 0;
  float sx = 0.0f, sy = 0.0f;
#pragma unroll
  for (int e = 0; e < 24; e++) {
    nv += msk[e] ? 1 : 0;
    sx += msk[e] ? vx[e] : 0.0f;
    sy += msk[e] ? vy[e] : 0.0f;
  }
  float den = (float)(nv < 1 ? 1 : nv);
  float mcx = sx / den, mcy = sy / den;

  // ---- angular keys: pseudoangle is strictly monotone in atan2 -> same order,
  //      no transcendentals (saves ~24 atan2 sequences per thread) ----
  float kk[24];
#pragma unroll
  for (int e = 0; e < 24; e++) {
    float rx = vx[e] - mcx, ry = vy[e] - mcy;
    float d = fabsf(rx) + fabsf(ry);
    float pa = (d > 0.0f) ? copysignf(1.0f - rx / d, ry) : 0.0f;
    kk[e] = msk[e] ? pa : 1.0e9f;
  }

  // ---- 9-pass min-extraction (register-resident; first-index wins on ties,
  //      matching jnp.argsort's stable order) + streaming shoelace ----
  float s0x = 0.0f, s0y = 0.0f, prx = 0.0f, pry = 0.0f, cross = 0.0f;
  for (int k = 0; k < 9; k++) {
    float m = 3.0e38f;
    float bx = 0.0f, by = 0.0f;
    int bi = 0;
#pragma unroll
    for (int e = 0; e < 24; e++) {
      bool c = kk[e] < m;
      m = c ? kk[e] : m;
      bx = c ? vx[e] : bx;
      by = c ? vy[e] : by;
      bi = c ? e : bi;
    }
#pragma unroll
    for (int e = 0; e < 24; e++) kk[e] = (e == bi) ? 3.0e38f : kk[e];
    if (k == 0) {
      s0x = bx; s0y = by; prx = bx; pry = by;
    } else {
      float sxk = (k < nv) ? bx : s0x;
      float syk = (k < nv) ? by : s0y;
      cross += prx * syk - pry * sxk;
      prx = sxk; pry = syk;
    }
  }
  float inter = fabsf(cross) * 0.5f;

  // ---- GIoU-style loss ----
  float area_p = p5[2] * p5[3];
  float area_g = g5[2] * g5[3];
  float uni = area_p + area_g - inter;
  float iou = inter / (uni + EPS);
  float en = ((tlx < brx) && (tly < bry)) ? 1.0f : 0.0f;
  float area_c = (brx - tlx) * (bry - tly) * en + EPS;
  float giou = iou - (area_c - uni) / area_c;
  float loss = 1.0f - fminf(fmaxf(giou, -1.0f), 1.0f);
  out[i] = loss;
}

extern "C" void kernel_launch(void* const* d_in, const int* in_sizes, int n_in,
                              void* d_out, int out_size, void* d_ws, size_t ws_size,
                              hipStream_t stream) {
  const float* pred = (const float*)d_in[0];
  const float* tgt = (const float*)d_in[1];
  float* out = (float*)d_out;
  int n = in_sizes[0] / 8;
  int blocks = (n + BLK - 1) / BLK;
  hipLaunchKernelGGL(piou_kernel, dim3(blocks), dim3(BLK), 0, stream,
                     pred, tgt, out, n);
}